// Classical_autoencoder_10746008174650
// MI455X (gfx1250) — compile-verified
//
#include <hip/hip_runtime.h>
#include <cstdint>

typedef __attribute__((ext_vector_type(2))) float v2f;
typedef __attribute__((ext_vector_type(8))) float v8f;

#define TILE   32
#define IMG_T  36            // tile + 4 halo rows/cols (Y0-2 .. Y0+33)
#define IMG_TP 37            // padded LDS row stride
#define NPY    17            // patch positions per tile dim (Y0-2..Y0+30 step 2)
#define NPATCH (NPY * NPY)   // 289
#define HH     1024
#define WW     1024

__global__ __launch_bounds__(256)
void ae_tile_kernel(const float* __restrict__ img,
                    const float* __restrict__ w_conv,
                    const float* __restrict__ b_conv,
                    const float* __restrict__ W_b,
                    const float* __restrict__ b_b,
                    const float* __restrict__ W_d,
                    const float* __restrict__ b_d,
                    float* __restrict__ out)
{
    __shared__ float simg[IMG_T * IMG_TP];           // image tile + halo
    __shared__ float sacc[TILE * (TILE + 1)];        // overlap-add accumulator
    __shared__ float szst[8][32][2];                 // per-wave WMMA D staging

    const int tid  = threadIdx.x;
    const int lane = tid & 31;
    const int wave = tid >> 5;
    const int X0   = blockIdx.x * TILE;
    const int Y0   = blockIdx.y * TILE;
    const int bz   = blockIdx.z;
    const float* imgB = img + (size_t)bz * HH * WW;

    // zero the accumulator
    for (int i = tid; i < TILE * (TILE + 1); i += 256) sacc[i] = 0.0f;

    // ---- async global -> LDS fill of the image tile (CDNA5 path) ----
    for (int i = tid; i < IMG_T * IMG_T; i += 256) {
        int r  = i / IMG_T, c = i % IMG_T;
        int gy = min(max(Y0 - 2 + r, 0), HH - 1);
        int gx = min(max(X0 - 2 + c, 0), WW - 1);
        uint32_t ldsb = (uint32_t)(uintptr_t)&simg[r * IMG_TP + c];
        uint32_t goff = (uint32_t)((gy * WW + gx) * 4);
        asm volatile("global_load_async_to_lds_b32 %0, %1, %2 offset:0"
                     :: "v"(ldsb), "v"(goff), "s"(imgB) : "memory");
    }
    asm volatile("s_wait_asynccnt 0" ::: "memory");
    __syncthreads();

    // uniform small weights (scalar-cached loads)
    float wcv[8], wdv[8];
#pragma unroll
    for (int i = 0; i < 8; ++i) { wcv[i] = w_conv[i]; wdv[i] = W_d[i]; }
    const float bc0 = b_conv[0], bc1 = b_conv[1];
    const float wb00 = W_b[0], wb01 = W_b[1], wb10 = W_b[2], wb11 = W_b[3];
    const float bb0 = b_b[0], bb1 = b_b[1];
    const float bd  = b_d[0];

    const int NG = (NPATCH + 31) / 32;   // 10 groups of 32 patches
    for (int g = wave; g < NG; g += 8) {
        int  p  = g * 32 + lane;
        bool pv = (p < NPATCH);
        int  pp = pv ? p : (NPATCH - 1);
        int  py = pp / NPY, px = pp % NPY;
        int  ly = 2 * py, lx = 2 * px;           // patch top-left in simg coords
        int  gy0 = Y0 - 2 + ly, gx0 = X0 - 2 + lx;
        bool valid = pv && gy0 >= 0 && gx0 >= 0 && gy0 <= HH - 4 && gx0 <= WW - 4;

        // load 4x4 patch from LDS
        float s[4][4];
#pragma unroll
        for (int r = 0; r < 4; ++r)
#pragma unroll
            for (int c = 0; c < 4; ++c)
                s[r][c] = simg[(ly + r) * IMG_TP + (lx + c)];

        // 2x2 conv (2ch) + ReLU + max/argmax over 4 blocks (first-max wins)
        float vals0, vals1; int idx0, idx1;
#pragma unroll
        for (int ch = 0; ch < 2; ++ch) {
            const float bcv = ch ? bc1 : bc0;
            float best = -1.0f; int bi = 0;
#pragma unroll
            for (int jk = 0; jk < 4; ++jk) {
                int j = jk >> 1, k = jk & 1;
                float v = bcv;
#pragma unroll
                for (int pq = 0; pq < 4; ++pq) {
                    int pr = pq >> 1, q = pq & 1;
                    v += s[2 * j + pr][2 * k + q] * wcv[ch * 4 + pr * 2 + q];
                }
                v = fmaxf(v, 0.0f);
                if (v > best) { best = v; bi = jk; }
            }
            if (ch == 0) { vals0 = best; idx0 = bi; }
            else         { vals1 = best; idx1 = bi; }
        }

        // ---- bottleneck matmul for the wave's 32 patches via WMMA ----
        // A (16x4 f32): lanes 0-15 hold M=patch, VGPR0=K0(ch0), VGPR1=K1(ch1);
        // lanes 16-31 (K2,K3) padded 0.  B (4x16): B[c][n] = W_b[n][c].
        float o0 = __shfl(vals0, (lane & 15) + 16, 32);
        float o1 = __shfl(vals1, (lane & 15) + 16, 32);
        bool lo = (lane < 16);
        v2f a0, a1, bm;
        a0.x = lo ? vals0 : 0.0f;  a0.y = lo ? vals1 : 0.0f;      // patches 0-15
        a1.x = lo ? o0    : 0.0f;  a1.y = lo ? o1    : 0.0f;      // patches 16-31
        bm.x = (lane == 0) ? wb00 : ((lane == 1) ? wb10 : 0.0f);  // B[0][n]
        bm.y = (lane == 0) ? wb01 : ((lane == 1) ? wb11 : 0.0f);  // B[1][n]
        v8f cz = {0.f, 0.f, 0.f, 0.f, 0.f, 0.f, 0.f, 0.f};
        v8f d0 = __builtin_amdgcn_wmma_f32_16x16x4_f32(false, a0, false, bm,
                                                       (short)0, cz, false, false);
        v8f d1 = __builtin_amdgcn_wmma_f32_16x16x4_f32(false, a1, false, bm,
                                                       (short)0, cz, false, false);

        // stage D back to patch-owning lanes through LDS
        {
            int n  = lane & 15;
            int mb = (lane & 16) ? 8 : 0;
            if (n < 2) {
#pragma unroll
                for (int r = 0; r < 8; ++r) {
                    szst[wave][mb + r][n]      = d0[r];
                    szst[wave][16 + mb + r][n] = d1[r];
                }
            }
        }
        __builtin_amdgcn_wave_barrier();
        asm volatile("s_wait_dscnt 0" ::: "memory");   // same-wave LDS in-order
        float z0 = fmaxf(szst[wave][lane][0] + bb0, 0.0f);
        float z1 = fmaxf(szst[wave][lane][1] + bb1, 0.0f);

        // unpool + 2x2 transposed conv + sigmoid, overlap-add into LDS tile
#pragma unroll
        for (int r = 0; r < 4; ++r) {
            int i2 = r >> 1, pr = r & 1;
#pragma unroll
            for (int c = 0; c < 4; ++c) {
                int j2 = c >> 1, q = c & 1;
                int bf = i2 * 2 + j2;
                float acc = bd;
                if (idx0 == bf) acc += z0 * wdv[pr * 2 + q];
                if (idx1 == bf) acc += z1 * wdv[4 + pr * 2 + q];
                float sg = 1.0f / (1.0f + __expf(-acc));
                int ty = ly - 2 + r;
                int tx = lx - 2 + c;
                if (valid && ty >= 0 && ty < TILE && tx >= 0 && tx < TILE)
                    atomicAdd(&sacc[ty * (TILE + 1) + tx], sg);   // ds_add_f32
            }
        }
    }
    __syncthreads();

    // normalize by analytic coverage count and store
    for (int i = tid; i < TILE * TILE; i += 256) {
        int ty = i >> 5, tx = i & 31;
        int gy = Y0 + ty, gx = X0 + tx;
        float cy = (gy >= 2 && gy <= HH - 3) ? 2.0f : 1.0f;
        float cx = (gx >= 2 && gx <= WW - 3) ? 2.0f : 1.0f;
        out[((size_t)bz * HH + gy) * WW + gx] = sacc[ty * (TILE + 1) + tx] / (cy * cx);
    }
}

extern "C" void kernel_launch(void* const* d_in, const int* in_sizes, int n_in,
                              void* d_out, int out_size, void* d_ws, size_t ws_size,
                              hipStream_t stream) {
    const float* img    = (const float*)d_in[0];
    const float* w_conv = (const float*)d_in[1];
    const float* b_conv = (const float*)d_in[2];
    const float* W_b    = (const float*)d_in[3];
    const float* b_b    = (const float*)d_in[4];
    const float* W_d    = (const float*)d_in[5];
    const float* b_d    = (const float*)d_in[6];
    float* out = (float*)d_out;

    int B = in_sizes[0] / (HH * WW);   // 8
    dim3 grid(WW / TILE, HH / TILE, B);
    ae_tile_kernel<<<grid, 256, 0, stream>>>(img, w_conv, b_conv, W_b, b_b, W_d, b_d, out);
}